// GateRecurrent2dnoind_90546500534704
// MI455X (gfx1250) — compile-verified
//
#include <hip/hip_runtime.h>

// GateRecurrent2dnoind (horizontal, forward), exact closed form of the
// 128-iteration reference via per-row linear-recurrence pair scans.
//
//   o_N[x] = I[x] + P[x] * s_{N-x}        (x <= N, N = 128)
//   o_N[x] = I[x]                          (x  > N; dropped terms ~ e^-90)
//   I,P    : inclusive scan of (a[x], b[x]) pairs under composition
//   s_m    : a0 * sum_{j<=m} g^j  (truncated geometric boundary series)
//
// Bandwidth-bound: 320 MiB total traffic -> ~14 us at 23.3 TB/s on MI455X.
// One wave32 per row; 8 contiguous columns per lane; ds_bpermute wave scans.

constexpr int WPB   = 8;          // waves (= rows) per block
constexpr int TPB   = WPB * 32;   // wave32
constexpr int WID   = 256;        // output width
constexpr int HGT   = 256;        // output height
constexpr int GW    = 128;        // gate width
constexpr int GH    = 128;        // gate height
constexpr int NITER = 128;        // max(h,w)//2 in the reference
constexpr int CPL   = 8;          // columns per lane (32*8 = 256)

struct WaveLds {
  float gy1[GW];        // y-blended upsample rows of G1
  float gy2[GW];        // y-blended upsample rows of G2
  float s[NITER + 4];   // s[0..128] boundary series (padded)
};

__global__ __launch_bounds__(TPB, 2)
void gate_scan_kernel(const float* __restrict__ X,
                      const float* __restrict__ G1,
                      const float* __restrict__ G2,
                      float* __restrict__ out) {
  __shared__ WaveLds sh[WPB];
  const int lane = threadIdx.x & 31;
  const int wid  = threadIdx.x >> 5;
  const int row  = blockIdx.x * WPB + wid;   // row = (b*64 + c)*256 + y
  WaveLds& ws = sh[wid];

  const int y     = row & (HGT - 1);
  const int plane = row >> 8;

  const float* __restrict__ xrow = X   + (size_t)row * WID;
  float*       __restrict__ orow = out + (size_t)row * WID;

  // CDNA5 global_prefetch_b8 path: warm the X row we are about to load.
  __builtin_prefetch(xrow + lane * CPL, 0, 3);

  // ---- stage 1: vertical bilinear blend of the two needed gate rows ----
  // half-pixel mapping: src_y = y*0.5 - 0.25 (matches jax bilinear resize)
  const float sy  = (float)y * 0.5f - 0.25f;
  const int   yf  = (int)floorf(sy);
  const float wy1 = sy - (float)yf;
  const float wy0 = 1.0f - wy1;
  const int y0 = yf < 0 ? 0 : yf;
  const int y1 = (yf + 1) > (GH - 1) ? (GH - 1) : (yf + 1);

  const float* g1p = G1 + (size_t)plane * (GH * GW);
  const float* g2p = G2 + (size_t)plane * (GH * GW);
  const int k4 = lane * 4;                       // 32 lanes * 4 = 128 cols
  {
    const float4 r0 = *(const float4*)(g1p + (size_t)y0 * GW + k4);
    const float4 r1 = *(const float4*)(g1p + (size_t)y1 * GW + k4);
    const float4 q0 = *(const float4*)(g2p + (size_t)y0 * GW + k4);
    const float4 q1 = *(const float4*)(g2p + (size_t)y1 * GW + k4);
    float4 m1, m2;
    m1.x = fmaf(wy0, r0.x, wy1 * r1.x);  m1.y = fmaf(wy0, r0.y, wy1 * r1.y);
    m1.z = fmaf(wy0, r0.z, wy1 * r1.z);  m1.w = fmaf(wy0, r0.w, wy1 * r1.w);
    m2.x = fmaf(wy0, q0.x, wy1 * q1.x);  m2.y = fmaf(wy0, q0.y, wy1 * q1.y);
    m2.z = fmaf(wy0, q0.z, wy1 * q1.z);  m2.w = fmaf(wy0, q0.w, wy1 * q1.w);
    *(float4*)(&ws.gy1[k4]) = m1;                // ds_store_b128
    *(float4*)(&ws.gy2[k4]) = m2;
  }

  // ---- load this lane's 8 contiguous X values ----
  float xv[CPL];
  {
    const float4 xa = *(const float4*)(xrow + lane * CPL);
    const float4 xb = *(const float4*)(xrow + lane * CPL + 4);
    xv[0] = xa.x; xv[1] = xa.y; xv[2] = xa.z; xv[3] = xa.w;
    xv[4] = xb.x; xv[5] = xb.y; xv[6] = xb.z; xv[7] = xb.w;
  }

  __syncthreads();   // gy1/gy2 visible to the whole wave

  // ---- stage 2: boundary series s_m = a0 * sum_{j<=m} g^j, m = 0..128 ----
  // Parallel scan of 128 identical pairs (a0, g): exact products, no pow(),
  // no 1/(1-g) cancellation, safe for g -> 0 and g -> 1.
  const float x0c = __shfl(xv[0], 0, 32);        // X[row][0] broadcast
  const float a0  = ws.gy1[0] * x0c;             // upsample at x=0 == gy[0]
  const float g   = ws.gy2[0];
  {
    float sA[4], sB[4];
    sA[0] = a0; sB[0] = g;
#pragma unroll
    for (int j = 1; j < 4; ++j) {
      sA[j] = fmaf(g, sA[j - 1], a0);
      sB[j] = g * sB[j - 1];
    }
    float A = sA[3], B = sB[3];
#pragma unroll
    for (int d = 1; d < 32; d <<= 1) {
      const float Au = __shfl_up(A, d, 32);
      const float Bu = __shfl_up(B, d, 32);
      if (lane >= d) { A = fmaf(B, Au, A); B *= Bu; }
    }
    float eA = __shfl_up(A, 1, 32);
    if (lane == 0) eA = 0.0f;
    float last = 0.0f;
#pragma unroll
    for (int j = 0; j < 4; ++j) {
      const float sv = fmaf(sB[j], eA, sA[j]);   // s_{4*lane+j}
      ws.s[4 * lane + j] = sv;
      last = sv;
    }
    if (lane == 31) ws.s[NITER] = fmaf(g, last, a0);  // s_128
  }

  // ---- stage 3: horizontal upsample from 6 contiguous taps ----
  // Lane l owns x = 8l..8l+7 -> needs gy[4l-1 .. 4l+4]:
  // one aligned b128 LDS load + two scalar taps per gate array.
  // gu[2t]   = 0.25*c[t-1] + 0.75*c[t]
  // gu[2t+1] = 0.75*c[t]   + 0.25*c[t+1]        (edges clamp-replicate)
  const int km1 = (k4 - 1 < 0) ? 0 : k4 - 1;
  const int kp4 = (k4 + 4 > GW - 1) ? GW - 1 : k4 + 4;
  float g1u[CPL], g2u[CPL];
  {
    const float4 c  = *(const float4*)(&ws.gy1[k4]);   // ds_load_b128
    const float cm  = ws.gy1[km1];
    const float cp  = ws.gy1[kp4];
    g1u[0] = fmaf(0.25f, cm,  0.75f * c.x);
    g1u[1] = fmaf(0.75f, c.x, 0.25f * c.y);
    g1u[2] = fmaf(0.25f, c.x, 0.75f * c.y);
    g1u[3] = fmaf(0.75f, c.y, 0.25f * c.z);
    g1u[4] = fmaf(0.25f, c.y, 0.75f * c.z);
    g1u[5] = fmaf(0.75f, c.z, 0.25f * c.w);
    g1u[6] = fmaf(0.25f, c.z, 0.75f * c.w);
    g1u[7] = fmaf(0.75f, c.w, 0.25f * cp);
  }
  {
    const float4 c  = *(const float4*)(&ws.gy2[k4]);
    const float cm  = ws.gy2[km1];
    const float cp  = ws.gy2[kp4];
    g2u[0] = fmaf(0.25f, cm,  0.75f * c.x);
    g2u[1] = fmaf(0.75f, c.x, 0.25f * c.y);
    g2u[2] = fmaf(0.25f, c.x, 0.75f * c.y);
    g2u[3] = fmaf(0.75f, c.y, 0.25f * c.z);
    g2u[4] = fmaf(0.25f, c.y, 0.75f * c.z);
    g2u[5] = fmaf(0.75f, c.z, 0.25f * c.w);
    g2u[6] = fmaf(0.25f, c.z, 0.75f * c.w);
    g2u[7] = fmaf(0.75f, c.w, 0.25f * cp);
  }

  // ---- stage 4: main pair scan over the 256 columns ----
  float LA[CPL], LB[CPL];                        // lane-local inclusive pairs
#pragma unroll
  for (int j = 0; j < CPL; ++j) {
    float a = g1u[j] * xv[j];
    float b = g2u[j];
    if (lane == 0 && j == 0) { a = 0.0f; b = 1.0f; }  // col 0 handled by s
    if (j == 0) { LA[0] = a; LB[0] = b; }
    else        { LA[j] = fmaf(b, LA[j - 1], a); LB[j] = b * LB[j - 1]; }
  }
  float A = LA[CPL - 1], B = LB[CPL - 1];
#pragma unroll
  for (int d = 1; d < 32; d <<= 1) {
    const float Au = __shfl_up(A, d, 32);
    const float Bu = __shfl_up(B, d, 32);
    if (lane >= d) { A = fmaf(B, Au, A); B *= Bu; }
  }
  float Aex = __shfl_up(A, 1, 32);
  float Bex = __shfl_up(B, 1, 32);
  if (lane == 0) { Aex = 0.0f; Bex = 1.0f; }

  __syncthreads();   // s[] visible before the apply phase reads it

  // ---- stage 5: apply exclusive prefix + boundary term, store ----
  float ov[CPL];
#pragma unroll
  for (int j = 0; j < CPL; ++j) {
    const int x = lane * CPL + j;
    const float I = fmaf(LB[j], Aex, LA[j]);     // interior scan value
    const float P = LB[j] * Bex;                 // product of b[1..x]
    const int m = NITER - x;
    float o = I;
    if (m >= 0) o = fmaf(P, ws.s[m], I);         // + P[x]*s_{128-x}
    ov[j] = o;                                   // x==0 -> 0 + 1*s_128  (exact)
  }
  float4 oA, oB;
  oA.x = ov[0]; oA.y = ov[1]; oA.z = ov[2]; oA.w = ov[3];
  oB.x = ov[4]; oB.y = ov[5]; oB.z = ov[6]; oB.w = ov[7];
  *(float4*)(orow + lane * CPL)     = oA;        // global_store_b128
  *(float4*)(orow + lane * CPL + 4) = oB;
}

extern "C" void kernel_launch(void* const* d_in, const int* in_sizes, int n_in,
                              void* d_out, int out_size, void* d_ws, size_t ws_size,
                              hipStream_t stream) {
  (void)n_in; (void)out_size; (void)d_ws; (void)ws_size;
  const float* X  = (const float*)d_in[0];
  const float* G1 = (const float*)d_in[1];
  const float* G2 = (const float*)d_in[2];
  // d_in[3] (G3) is resized in the original model but never used.
  float* out = (float*)d_out;

  const int rows   = in_sizes[0] / WID;          // 8*64*256 = 131072
  const int blocks = rows / WPB;                 // 16384, exact
  gate_scan_kernel<<<blocks, TPB, 0, stream>>>(X, G1, G2, out);
}